// News_User_co_Encoder_23313082483404
// MI455X (gfx1250) — compile-verified
//
#include <hip/hip_runtime.h>
#include <hip/hip_bf16.h>

// ---------------------------------------------------------------------------
// Mathematically exact reduction of the reference:
// softmax over singleton axes forces news_att == user_att == 1, so
//   out[b] = dot( LN(colsum(news[b])), LN(colsum(user[b])) )
// 164 MB of f32 reads -> pure HBM-bandwidth problem (~7 us at 23.3 TB/s).
// Column sums over N=100 are done with V_WMMA_F32_16X16X4_F32 using an
// all-ones A matrix (free on the bandwidth roofline; exercises the CDNA5
// matrix path with full f32 precision).
// ---------------------------------------------------------------------------

#define B_SZ 512
#define N_SZ 100
#define D_SZ 400
#define EPSF 1e-5f

typedef __attribute__((ext_vector_type(2))) float v2f;
typedef __attribute__((ext_vector_type(8))) float v8f;

#ifndef __has_builtin
#define __has_builtin(x) 0
#endif
#if __has_builtin(__builtin_amdgcn_wmma_f32_16x16x4_f32)
#define USE_WMMA_F32X4 1
#else
#define USE_WMMA_F32X4 0
#endif

// Block-wide reduction of a float2 across 256 threads (result broadcast).
__device__ __forceinline__ float2 block_reduce2(float2 v, float2* red) {
    const int t = threadIdx.x;
    red[t] = v;
    __syncthreads();
#pragma unroll
    for (int s = 128; s > 0; s >>= 1) {
        if (t < s) {
            float2 o = red[t + s];
            float2 m = red[t];
            m.x += o.x;
            m.y += o.y;
            red[t] = m;
        }
        __syncthreads();
    }
    float2 r = red[0];
    __syncthreads();
    return r;
}

__global__ __launch_bounds__(256)
void fused_pool_ln_dot(const float* __restrict__ news,
                       const float* __restrict__ user,
                       const float* __restrict__ g1, const float* __restrict__ b1,
                       const float* __restrict__ g2, const float* __restrict__ b2,
                       float* __restrict__ out)
{
    __shared__ float  sums[2][D_SZ];   // [news|user] column sums over N
    __shared__ float2 red[256];

    const int b    = blockIdx.x;
    const int t    = threadIdx.x;
    const int wave = t >> 5;           // 0..7 (wave32)
    const int lane = t & 31;
    const int half = lane >> 4;        // 0: lanes 0-15, 1: lanes 16-31
    const int lcol = lane & 15;        // column within 16-wide d-tile

    const float* newsb = news + (size_t)b * (N_SZ * D_SZ);
    const float* userb = user + (size_t)b * (N_SZ * D_SZ);

    // ------------------------------------------------------------------
    // Phase 1: column sums over N=100 for both matrices.
    // 50 wave-tasks per block: task = mat*25 + dtile_index (16 cols each).
    // WMMA: D = ones(16x4) x B(4x16) + C accumulates 4 rows per step;
    // with all-ones A, the K-slot permutation inside the B fragment is
    // irrelevant (all K are summed), only the lane->N mapping matters.
    // Each half-wave stages 2 rows x 16 cols (64B-aligned segments).
    // ------------------------------------------------------------------
    for (int task = wave; task < 50; task += 8) {
        const int mat   = (task >= 25) ? 1 : 0;
        const int dtile = (task - mat * 25) << 4;
        const float* src = mat ? userb : newsb;
        const float* p = src + (2 * half) * D_SZ + dtile + lcol;

#if USE_WMMA_F32X4
        v8f c = {};
        v2f a;
        a[0] = 1.0f;
        a[1] = 1.0f;
#pragma unroll
        for (int k = 0; k < 25; ++k) {
            v2f bm;
            bm[0] = p[0];        // row n0 + 2*half
            bm[1] = p[D_SZ];     // row n0 + 2*half + 1
            p += 4 * D_SZ;
            // 8 args: (neg_a, A, neg_b, B, c_mod, C, reuse_a, reuse_b)
            c = __builtin_amdgcn_wmma_f32_16x16x4_f32(
                    false, a, false, bm, (short)0, c, false, false);
        }
        // C/D layout: VGPR0, lanes 0-15 hold row M=0 (the column sums).
        if (lane < 16) sums[mat][dtile + lcol] = c[0];
#else
        float acc = 0.0f;
#pragma unroll
        for (int k = 0; k < 25; ++k) {
            acc += p[0] + p[D_SZ];
            p += 4 * D_SZ;
        }
        acc += __shfl_down(acc, 16, 32);   // combine the two half-waves
        if (lane < 16) sums[mat][dtile + lcol] = acc;
#endif
    }
    __syncthreads();

    // ------------------------------------------------------------------
    // Phase 2: LayerNorm(sums) for both, then dot product over D=400.
    // Faithful two-pass variance: var = mean((x - mu)^2).
    // Thread t covers d = t and (if t < 144) d = t + 256.
    // ------------------------------------------------------------------
    const bool hi = (t < (D_SZ - 256));    // t < 144
    const float xn1 = sums[0][t];
    const float xu1 = sums[1][t];
    const float xn2 = hi ? sums[0][t + 256] : 0.0f;
    const float xu2 = hi ? sums[1][t + 256] : 0.0f;

    const float inv_d = 1.0f / (float)D_SZ;

    float2 s1 = block_reduce2(make_float2(xn1 + xn2, xu1 + xu2), red);
    const float mun = s1.x * inv_d;
    const float muu = s1.y * inv_d;

    const float dn1 = xn1 - mun, du1 = xu1 - muu;
    const float dn2 = hi ? (xn2 - mun) : 0.0f;
    const float du2 = hi ? (xu2 - muu) : 0.0f;

    float2 s2 = block_reduce2(
        make_float2(dn1 * dn1 + dn2 * dn2, du1 * du1 + du2 * du2), red);
    const float vn = s2.x * inv_d + EPSF;
    const float vu = s2.y * inv_d + EPSF;

    // rsqrt + one Newton step (matches jax.lax.rsqrt to f32 accuracy)
    float rn = rsqrtf(vn); rn = rn * (1.5f - 0.5f * vn * rn * rn);
    float ru = rsqrtf(vu); ru = ru * (1.5f - 0.5f * vu * ru * ru);

    float dot = (dn1 * rn * g1[t] + b1[t]) * (du1 * ru * g2[t] + b2[t]);
    if (hi) {
        const float yn2 = dn2 * rn * g1[t + 256] + b1[t + 256];
        const float yu2 = du2 * ru * g2[t + 256] + b2[t + 256];
        dot += yn2 * yu2;
    }

    float2 s3 = block_reduce2(make_float2(dot, 0.0f), red);
    if (t == 0) out[b] = s3.x;
}

extern "C" void kernel_launch(void* const* d_in, const int* in_sizes, int n_in,
                              void* d_out, int out_size, void* d_ws, size_t ws_size,
                              hipStream_t stream) {
    (void)in_sizes; (void)n_in; (void)out_size; (void)d_ws; (void)ws_size;
    const float* news = (const float*)d_in[0];
    const float* user = (const float*)d_in[1];
    // d_in[2..11] (attention MLP weights) are mathematically dead: the
    // reference applies softmax over singleton axes, collapsing both
    // attention vectors to exact ones.
    const float* g1 = (const float*)d_in[12];
    const float* b1 = (const float*)d_in[13];
    const float* g2 = (const float*)d_in[14];
    const float* b2 = (const float*)d_in[15];
    float* out = (float*)d_out;

    fused_pool_ln_dot<<<B_SZ, 256, 0, stream>>>(news, user, g1, b1, g2, b2, out);
}